// CenterPool_18545668784867
// MI455X (gfx1250) — compile-verified
//
#include <hip/hip_runtime.h>

typedef float v2f __attribute__((ext_vector_type(2)));
typedef float v8f __attribute__((ext_vector_type(8)));

#define IMG_WF 512.0f
#define FM 64
#define C_DIM 256

// One wave (32 lanes) computes one 16x16 tile of out (512 x 256):
//   out[p][c] = input[p>>4, c, cy(p), cx(p)] + label(p) . Wt[:,c] + bias[c]
// via V_WMMA_F32_16X16X4_F32 (K=4 == label dim, exact fit, full f32).
__global__ __launch_bounds__(256) void centerpool_wmma_kernel(
    const float* __restrict__ inp,     // (32, 256, 64, 64)
    const float* __restrict__ bboxes,  // (512, 4)
    const float* __restrict__ Wm,      // (256, 4)
    const float* __restrict__ bias,    // (256,)
    float* __restrict__ out)           // (512, 256)
{
    const int lane = threadIdx.x & 31;
    const int wave = threadIdx.x >> 5;
    const int tile = blockIdx.x * 8 + wave;   // 0..511
    const int rt   = tile >> 4;               // row tile  0..31  (16 points each)
    const int ct   = tile & 15;               // col tile  0..15  (16 channels each)

    const int half = lane >> 4;               // 0: K/M low half, 1: high half
    const int r    = lane & 15;               // row within tile decoded by this lane
    const int p    = rt * 16 + r;             // global point index 0..511
    const int c    = ct * 16 + (lane & 15);   // channel column for B/C/D

    // ---- decode bbox for row r (both halves redundantly decode, cheap) ----
    const float4 bb = ((const float4*)bboxes)[p];
    const float cw = IMG_WF / (float)FM;      // 8.0
    float img_xc = bb.x + floorf(bb.z * 0.5f);
    float img_yc = bb.y + floorf(bb.w * 0.5f);
    int cell_x = (int)floorf(img_xc / cw);
    int cell_y = (int)floorf(img_yc / cw);
    cell_x = min(max(cell_x, 0), FM - 1);
    cell_y = min(max(cell_y, 0), FM - 1);
    float lx = (img_xc - (float)cell_x * cw) / cw;
    float ly = (img_yc - (float)cell_y * cw) / cw;
    float lw = bb.z / IMG_WF;
    float lh = bb.w / IMG_WF;

    int xyoff = cell_y * FM + cell_x;         // offset within one (64,64) plane

    // ---- A matrix 16x4 f32 (2 VGPRs): lane -> row M=r, K = 2*half, 2*half+1 ----
    v2f a;
    a.x = half ? lw : lx;   // K = 0 or 2
    a.y = half ? lh : ly;   // K = 1 or 3

    // ---- B matrix 4x16 f32 (2 VGPRs): lane -> col c, K = 2*half, 2*half+1 ----
    // Wt[k][c] = W[c][k]; W is (256,4) row-major.
    v2f bm;
    bm.x = Wm[c * 4 + 2 * half];
    bm.y = Wm[c * 4 + 2 * half + 1];

    const float bv = bias[c];

    // ---- C accumulator: VGPR v holds row M = v + 8*half, column c ----
    v8f acc;
#pragma unroll
    for (int v = 0; v < 8; ++v) {
        int M  = v + 8 * half;
        int xo = __shfl(xyoff, M, 32);        // gather offset of row M
        int pm = rt * 16 + M;
        int img = pm >> 4;                    // p / N, N == 16
        acc[v] = inp[((size_t)img * C_DIM + c) * (size_t)(FM * FM) + xo] + bv;
    }

    // ---- D = A x B + C  (single f32 WMMA, K=4) ----
    v8f d = __builtin_amdgcn_wmma_f32_16x16x4_f32(
        false, a, false, bm, (short)0, acc, false, false);

    // ---- store 16x16 tile ----
#pragma unroll
    for (int v = 0; v < 8; ++v) {
        int pm = rt * 16 + v + 8 * half;
        out[(size_t)pm * C_DIM + c] = d[v];
    }
}

extern "C" void kernel_launch(void* const* d_in, const int* in_sizes, int n_in,
                              void* d_out, int out_size, void* d_ws, size_t ws_size,
                              hipStream_t stream) {
    const float* inp    = (const float*)d_in[0];  // (32,256,64,64)
    const float* bboxes = (const float*)d_in[1];  // (8,4,16,4)
    const float* Wm     = (const float*)d_in[2];  // (256,4)
    const float* bias   = (const float*)d_in[3];  // (256,)
    float* out = (float*)d_out;                   // (8,4,16,256) = 512x256

    // 512 tiles, 8 waves (256 threads) per block -> 64 blocks
    centerpool_wmma_kernel<<<64, 256, 0, stream>>>(inp, bboxes, Wm, bias, out);
}